// GCN_15204184228222
// MI455X (gfx1250) — compile-verified
//
#include <hip/hip_runtime.h>
#include <hip/hip_bf16.h>

typedef __attribute__((ext_vector_type(2))) float v2f;
typedef __attribute__((ext_vector_type(8))) float v8f;

#define F_IN 512
#define HID  16
#define CLS  7

// ---------------- degree / normalization ----------------
__global__ void deg_init_k(float* __restrict__ deg, int n) {
    int i = blockIdx.x * blockDim.x + threadIdx.x;
    if (i < n) deg[i] = 1.0f;               // self-loop contributes 1
}

__global__ void deg_count_k(const int* __restrict__ dst, float* __restrict__ deg, int E) {
    int e = blockIdx.x * blockDim.x + threadIdx.x;
    if (e < E) unsafeAtomicAdd(&deg[dst[e]], 1.0f);
}

__global__ void dinv_k(float* __restrict__ deg, int n) {
    int i = blockIdx.x * blockDim.x + threadIdx.x;
    if (i < n) deg[i] = rsqrtf(deg[i]);     // deg >= 1 always
}

// ---------------- GEMM1: h = x @ W1  (WMMA f32 16x16x4) ----------------
__global__ void __launch_bounds__(256) gemm1_k(const float* __restrict__ x,
                                               const float* __restrict__ W1,
                                               float* __restrict__ h, int nTiles) {
    __shared__ float sW[F_IN * HID];        // 32 KB, L2->LDS once per block
    for (int i = threadIdx.x * 4; i < F_IN * HID; i += 256 * 4)
        *(float4*)(sW + i) = *(const float4*)(W1 + i);
    __syncthreads();

    const int wave = threadIdx.x >> 5;
    const int lane = threadIdx.x & 31;
    const int tile = blockIdx.x * 8 + wave;
    if (tile >= nTiles) return;             // wave-uniform: EXEC stays all-1s

    const int m    = lane & 15;             // A row / B col / D col
    const int kk   = (lane >> 4) << 1;      // K sub-pair for this half-wave
    const int half = lane >> 4;

    const float* arow = x + (size_t)(tile * 16 + m) * F_IN + kk;
    v8f c = {0.f, 0.f, 0.f, 0.f, 0.f, 0.f, 0.f, 0.f};

#pragma unroll 4
    for (int k0 = 0; k0 < F_IN; k0 += 4) {
        v2f a = *(const v2f*)(arow + k0);                 // global_load_b64
        v2f b;
        b.x = sW[(k0 + kk)     * HID + m];                // ds_load_b32
        b.y = sW[(k0 + kk + 1) * HID + m];
        c = __builtin_amdgcn_wmma_f32_16x16x4_f32(false, a, false, b,
                                                  (short)0, c, false, false);
    }

    float* orow = h + (size_t)(tile * 16 + half * 8) * HID + m;
#pragma unroll
    for (int v = 0; v < 8; ++v) orow[(size_t)v * HID] = c[v];
}

// ---------------- layer-1 aggregation ----------------
__global__ void self_init1_k(const float* __restrict__ h, const float* __restrict__ dinv,
                             float* __restrict__ agg, int n) {
    int t = blockIdx.x * blockDim.x + threadIdx.x;      // n = N*HID
    if (t >= n) return;
    int i = t >> 4;
    float di = dinv[i];
    agg[t] = di * di * h[t];
}

__global__ void scatter1_k(const int* __restrict__ src, const int* __restrict__ dst,
                           const float* __restrict__ dinv, const float* __restrict__ h,
                           float* __restrict__ agg, int E) {
    int t = blockIdx.x * blockDim.x + threadIdx.x;      // E*16 threads
    int e = t >> 4;
    if (e >= E) return;
    int f = t & 15;
    int s = src[e], d = dst[e];
    float w = dinv[s] * dinv[d];
    unsafeAtomicAdd(&agg[(size_t)d * HID + f], w * h[(size_t)s * HID + f]);
}

__global__ void bias_relu_k(float* __restrict__ agg, const float* __restrict__ b1, int n) {
    int t = blockIdx.x * blockDim.x + threadIdx.x;
    if (t >= n) return;
    float v = agg[t] + b1[t & 15];
    agg[t] = v > 0.f ? v : 0.f;
}

// ---------------- GEMM2: h2 = relu @ W2  (WMMA, W2 zero-padded to 16x16) ---------
__global__ void __launch_bounds__(256) gemm2_k(const float* __restrict__ a_in,
                                               const float* __restrict__ W2,
                                               float* __restrict__ h2, int nTiles) {
    __shared__ float sW[HID * 16];          // 16x16, cols >= CLS are zero
    {
        int t = threadIdx.x;                // 256 == 16*16
        int k = t >> 4, n = t & 15;
        sW[t] = (n < CLS) ? W2[k * CLS + n] : 0.0f;
    }
    __syncthreads();

    const int wave = threadIdx.x >> 5;
    const int lane = threadIdx.x & 31;
    const int tile = blockIdx.x * 8 + wave;
    if (tile >= nTiles) return;

    const int m    = lane & 15;
    const int kk   = (lane >> 4) << 1;
    const int half = lane >> 4;

    const float* arow = a_in + (size_t)(tile * 16 + m) * HID + kk;
    v8f c = {0.f, 0.f, 0.f, 0.f, 0.f, 0.f, 0.f, 0.f};

#pragma unroll
    for (int k0 = 0; k0 < HID; k0 += 4) {
        v2f a = *(const v2f*)(arow + k0);
        v2f b;
        b.x = sW[(k0 + kk)     * 16 + m];
        b.y = sW[(k0 + kk + 1) * 16 + m];
        c = __builtin_amdgcn_wmma_f32_16x16x4_f32(false, a, false, b,
                                                  (short)0, c, false, false);
    }

    if (m < CLS) {
#pragma unroll
        for (int v = 0; v < 8; ++v) {
            int row = tile * 16 + half * 8 + v;
            h2[(size_t)row * CLS + m] = c[v];
        }
    }
}

// ---------------- layer-2 aggregation ----------------
__global__ void self_init2_k(const float* __restrict__ h2, const float* __restrict__ dinv,
                             const float* __restrict__ b2, float* __restrict__ out, int n) {
    int t = blockIdx.x * blockDim.x + threadIdx.x;      // n = N*CLS
    if (t >= n) return;
    int i = t / CLS, f = t % CLS;
    float di = dinv[i];
    out[t] = di * di * h2[t] + b2[f];
}

__global__ void scatter2_k(const int* __restrict__ src, const int* __restrict__ dst,
                           const float* __restrict__ dinv, const float* __restrict__ h2,
                           float* __restrict__ out, int E) {
    int t = blockIdx.x * blockDim.x + threadIdx.x;      // E*8 threads
    int e = t >> 3;
    if (e >= E) return;
    int f = t & 7;
    if (f >= CLS) return;
    int s = src[e], d = dst[e];
    float w = dinv[s] * dinv[d];
    unsafeAtomicAdd(&out[(size_t)d * CLS + f], w * h2[(size_t)s * CLS + f]);
}

// ---------------- launcher ----------------
extern "C" void kernel_launch(void* const* d_in, const int* in_sizes, int n_in,
                              void* d_out, int out_size, void* d_ws, size_t ws_size,
                              hipStream_t stream) {
    const float* x  = (const float*)d_in[0];
    const int*   ei = (const int*)  d_in[1];
    const float* W1 = (const float*)d_in[2];
    const float* b1 = (const float*)d_in[3];
    const float* W2 = (const float*)d_in[4];
    const float* b2 = (const float*)d_in[5];
    float* out = (float*)d_out;

    const int N = in_sizes[0] / F_IN;       // 100000
    const int E = in_sizes[1] / 2;          // 3200000
    const int* src = ei;
    const int* dst = ei + E;

    // workspace: dinv[N] | h[N*16] | agg[N*16] | h2[N*7]   (~16 MB)
    float* dinv = (float*)d_ws;
    float* h    = dinv + N;
    float* agg  = h   + (size_t)N * HID;
    float* h2   = agg + (size_t)N * HID;

    const int B = 256;
    const int nTiles = N / 16;              // 6250, exact

    deg_init_k <<<(N + B - 1) / B, B, 0, stream>>>(dinv, N);
    deg_count_k<<<(E + B - 1) / B, B, 0, stream>>>(dst, dinv, E);
    dinv_k     <<<(N + B - 1) / B, B, 0, stream>>>(dinv, N);

    gemm1_k    <<<(nTiles + 7) / 8, B, 0, stream>>>(x, W1, h, nTiles);

    int n1 = N * HID;
    self_init1_k<<<(n1 + B - 1) / B, B, 0, stream>>>(h, dinv, agg, n1);
    {
        long long tot = (long long)E * 16;
        scatter1_k<<<(int)((tot + B - 1) / B), B, 0, stream>>>(src, dst, dinv, h, agg, E);
    }
    bias_relu_k<<<(n1 + B - 1) / B, B, 0, stream>>>(agg, b1, n1);

    gemm2_k    <<<(nTiles + 7) / 8, B, 0, stream>>>(agg, W2, h2, nTiles);

    int n2 = N * CLS;
    self_init2_k<<<(n2 + B - 1) / B, B, 0, stream>>>(h2, dinv, b2, out, n2);
    {
        long long tot = (long long)E * 8;
        scatter2_k<<<(int)((tot + B - 1) / B), B, 0, stream>>>(src, dst, dinv, h2, out, E);
    }
}